// PatchShuffle_82746839924893
// MI455X (gfx1250) — compile-verified
//
#include <hip/hip_runtime.h>
#include <stdint.h>

#define N_ROWS 1024   // 64 * 16 (B*C)
#define S_LEN  1200
#define KEEP   480    // 1200 - int(18/30*1200)
#define HALF   614400 // N_ROWS*S_LEN/2 : threefry counter split point

__device__ __forceinline__ uint32_t rotl32(uint32_t x, int r) {
  return (x << r) | (x >> (32 - r));
}

// JAX Threefry-2x32, key = (0, 42)  (jax.random.key(42)), 20 rounds.
__device__ __forceinline__ void threefry2x32_0_42(uint32_t x0, uint32_t x1,
                                                  uint32_t& o0, uint32_t& o1) {
  const uint32_t ks0 = 0u;
  const uint32_t ks1 = 42u;
  const uint32_t ks2 = 0u ^ 42u ^ 0x1BD11BDAu;
  x0 += ks0; x1 += ks1;
#define TF_R(r) { x0 += x1; x1 = rotl32(x1, (r)); x1 ^= x0; }
  TF_R(13) TF_R(15) TF_R(26) TF_R(6)
  x0 += ks1; x1 += ks2 + 1u;
  TF_R(17) TF_R(29) TF_R(16) TF_R(24)
  x0 += ks2; x1 += ks0 + 2u;
  TF_R(13) TF_R(15) TF_R(26) TF_R(6)
  x0 += ks0; x1 += ks1 + 3u;
  TF_R(17) TF_R(29) TF_R(16) TF_R(24)
  x0 += ks1; x1 += ks2 + 4u;
  TF_R(13) TF_R(15) TF_R(26) TF_R(6)
  x0 += ks2; x1 += ks0 + 5u;
#undef TF_R
  o0 = x0; o1 = x1;
}

// One block handles rows r and r+512 (they share threefry counter pairs
// (n, n+HALF)).  uniform() is monotone in the 23-bit field (bits>>9), so
// stable selection of the 480 smallest == radix selection on those 23 bits
// with index tie-break.  O(N) per row via a 4096-bin histogram on the top
// 12 bits; the (rare) threshold-bin ties are resolved exactly with a tiny
// packed-key list.
__global__ __launch_bounds__(256)
void PatchShuffle_select_kernel(int* __restrict__ keep_src,
                                float* __restrict__ idx_out) {
  __shared__ uint32_t keyA[S_LEN];
  __shared__ uint32_t keyB[S_LEN];
  __shared__ int hist[4096];
  __shared__ int chunksum[256];
  __shared__ uint32_t binlist[S_LEN];   // packed (low11<<16)|idx for bin T
  __shared__ int keptpos[S_LEN];        // output slot if kept, else -1
  __shared__ int sT, sNeed, sListCnt;

  const int tid = threadIdx.x;
  const int r   = blockIdx.x;           // 0..511 -> rows r and r+512

  for (int s = tid; s < S_LEN; s += 256) {
    uint32_t n = (uint32_t)(r * S_LEN + s);
    uint32_t o0, o1;
    threefry2x32_0_42(n, n + HALF, o0, o1);
    keyA[s] = o0 >> 9;                  // 23-bit order key, ties = float ties
    keyB[s] = o1 >> 9;
  }
  __syncthreads();

  for (int rr = 0; rr < 2; ++rr) {
    const uint32_t* key = rr ? keyB : keyA;
    const int row = r + rr * 512;

    // --- histogram of top 12 bits ---
    for (int b = tid; b < 4096; b += 256) hist[b] = 0;
    if (tid == 0) sListCnt = 0;
    __syncthreads();
    for (int s = tid; s < S_LEN; s += 256)
      atomicAdd(&hist[key[s] >> 11], 1);
    __syncthreads();

    // --- exclusive prefix over 4096 bins (16 bins / thread) ---
    {
      int base = tid * 16;
      int sum = 0;
      for (int k = 0; k < 16; ++k) sum += hist[base + k];
      chunksum[tid] = sum;
    }
    __syncthreads();
    if (tid == 0) {
      int c = 0;
      for (int i = 0; i < 256; ++i) { int v = chunksum[i]; chunksum[i] = c; c += v; }
    }
    __syncthreads();
    {
      int base = tid * 16;
      int c = chunksum[tid];                       // exclusive prefix at chunk start
      for (int k = 0; k < 16; ++k) {
        int h = hist[base + k];
        if (c < KEEP && KEEP <= c + h) {           // exactly one bin matches
          sT = base + k;
          sNeed = KEEP - c;                        // how many to keep from bin T
        }
        c += h;
      }
    }
    __syncthreads();
    const int T = sT;
    const int need = sNeed;

    // --- collect bin-T members as packed (low11, index) ---
    for (int s = tid; s < S_LEN; s += 256) {
      if ((int)(key[s] >> 11) == T) {
        int p = atomicAdd(&sListCnt, 1);
        binlist[p] = ((key[s] & 0x7FFu) << 16) | (uint32_t)s;
      }
    }
    __syncthreads();
    const int lcnt = sListCnt;

    // --- kept flags (stable: packed compare == (noise, index) lexicographic) ---
    for (int s = tid; s < S_LEN; s += 256) {
      int t = (int)(key[s] >> 11);
      int kept;
      if (t < T)      kept = 1;
      else if (t > T) kept = 0;
      else {
        uint32_t me = ((key[s] & 0x7FFu) << 16) | (uint32_t)s;
        int rnk = 0;
        for (int i = 0; i < lcnt; ++i) rnk += (binlist[i] < me);
        kept = (rnk < need);
      }
      keptpos[s] = kept;
    }
    __syncthreads();

    // --- ascending-index output slots (short serial scan) ---
    if (tid == 0) {
      int c = 0;
      for (int s = 0; s < S_LEN; ++s) {
        int kf = keptpos[s];
        keptpos[s] = kf ? c : -1;
        c += kf;
      }
    }
    __syncthreads();

    for (int s = tid; s < S_LEN; s += 256) {
      int p = keptpos[s];
      if (p >= 0) {
        int o = row * KEEP + p;
        keep_src[o] = s;
        idx_out[o]  = (float)s;         // keep_idx tuple output
      }
    }
    __syncthreads();                    // hist/lists reused for row B
  }
}

// Bandwidth-bound row gather via the CDNA5 async copy engine:
// GLOBAL_LOAD_ASYNC_TO_LDS_B128 -> s_wait_asynccnt -> GLOBAL_STORE_ASYNC_FROM_LDS_B128.
// One 16-byte vector per lane; a 256-thread block moves 4 contiguous 1 KB rows.
__global__ __launch_bounds__(256)
void PatchShuffle_gather_kernel(const float4* __restrict__ in4,
                                const int* __restrict__ keep_src,
                                float4* __restrict__ out4) {
  __shared__ float4 stage[256];

  const unsigned t   = blockIdx.x * 256u + threadIdx.x;  // float4 index, total 31,457,280
  const unsigned k   = t >> 6;                           // output row (64 float4 per row)
  const unsigned off = t & 63u;
  const unsigned bc  = k / KEEP;                         // (b*16+c)
  const unsigned s   = (unsigned)keep_src[k];            // kept source patch
  const unsigned src = (bc * S_LEN + s) * 64u + off;

  unsigned ldsoff = (unsigned)(size_t)&stage[threadIdx.x];            // LDS byte offset
  unsigned long long gsrc = (unsigned long long)(size_t)(in4 + src);
  unsigned long long gdst = (unsigned long long)(size_t)(out4 + t);

  asm volatile("global_load_async_to_lds_b128 %0, %1, off"
               :: "v"(ldsoff), "v"(gsrc) : "memory");
  asm volatile("s_wait_asynccnt 0" ::: "memory");
  asm volatile("global_store_async_from_lds_b128 %0, %1, off"
               :: "v"(gdst), "v"(ldsoff) : "memory");
  // s_endpgm performs an implicit wait-idle, covering the outstanding store.
}

extern "C" void kernel_launch(void* const* d_in, const int* in_sizes, int n_in,
                              void* d_out, int out_size, void* d_ws, size_t ws_size,
                              hipStream_t stream) {
  (void)in_sizes; (void)n_in; (void)out_size; (void)ws_size;

  const float* patches = (const float*)d_in[0];
  float* out = (float*)d_out;
  // tuple layout: unmasked (64*16*480*256 floats) then keep_idx (64*16*480)
  float* idx_out = out + (size_t)N_ROWS * KEEP * 256;   // offset 125,829,120
  int* keep_src = (int*)d_ws;                           // 1024*480 ints scratch

  PatchShuffle_select_kernel<<<512, 256, 0, stream>>>(keep_src, idx_out);

  const unsigned total_vec = (unsigned)N_ROWS * KEEP * 64u;   // 31,457,280 float4
  PatchShuffle_gather_kernel<<<total_vec / 256u, 256, 0, stream>>>(
      (const float4*)patches, keep_src, (float4*)out);
}